// AtomicConv_47923245089375
// MI455X (gfx1250) — compile-verified
//
#include <hip/hip_runtime.h>

// AtomicConv symmetry-function kernel for gfx1250 (MI455X, wave32).
// Per (b,n): sym[l,a] = sum_m rsf[l,m] * mask[m,a]  ==  (16x64)@(64x4) matmul,
// mapped onto 16 chained V_WMMA_F32_16X16X4_F32 ops (exact f32 accumulate).

typedef __attribute__((ext_vector_type(2))) float v2f;
typedef __attribute__((ext_vector_type(8))) float v8f;

#define B_  32
#define N_  2048
#define M_  64
#define L_  16
#define A_  4
#define WAVES_PER_BLOCK 8
#define PI_F 3.14159265358979323846f

__global__ __launch_bounds__(256) void atomic_conv_wmma(
    const float* __restrict__ X,          // (B, N, 3)
    const int*   __restrict__ Nbrs,       // (B, N, M)
    const int*   __restrict__ NbrsZ,      // (B, N, M)
    const float* __restrict__ rc,         // (L,)
    const float* __restrict__ rs,         // (L,)
    const float* __restrict__ e,          // (L,)
    const int*   __restrict__ atom_types, // (A,)
    float*       __restrict__ out)        // (L*A, B, N)
{
    // Per-wave LDS slices: shared R and Z for the 64 neighbors of this (b,n).
    __shared__ float sR[WAVES_PER_BLOCK][M_];
    __shared__ int   sZ[WAVES_PER_BLOCK][M_];

    const int tid  = threadIdx.x;
    const int w    = tid >> 5;       // wave in block
    const int lane = tid & 31;
    const int gw   = blockIdx.x * WAVES_PER_BLOCK + w;   // 0 .. B*N-1 exactly
    const int b    = gw >> 11;                           // / N_
    const int n    = gw & (N_ - 1);

    // ---- Phase 1: lane t computes R, Z for neighbors m = 2t, 2t+1 ----
    const int   base_nm = b * N_ + n;
    const float cx = X[base_nm * 3 + 0];
    const float cy = X[base_nm * 3 + 1];
    const float cz = X[base_nm * 3 + 2];

    const int m0 = lane * 2;
#pragma unroll
    for (int q = 0; q < 2; ++q) {
        const int m   = m0 + q;
        const int idx = Nbrs[base_nm * M_ + m];
        const int xb  = (b * N_ + idx) * 3;
        const float dx = X[xb + 0] - cx;
        const float dy = X[xb + 1] - cy;
        const float dz = X[xb + 2] - cz;
        sR[w][m] = sqrtf(dx * dx + dy * dy + dz * dz);
        sZ[w][m] = NbrsZ[base_nm * M_ + m];
    }
    __syncthreads();

    // ---- Phase 2: 16 chained f32 WMMAs over K = m (4 per op) ----
    // A (16x4 f32) layout: lane t holds row l = t&15; lanes 0-15 -> K=0,1,
    // lanes 16-31 -> K=2,3.  B (4x16 f32): lane t holds col a = t&15 with the
    // same K split.  So each lane only ever needs rsf for ONE filter l and the
    // mask for ONE atom-type column a.
    const int   row  = lane & 15;   // l for A-rows; a-col for B and D
    const int   hi   = lane >> 4;   // selects K pair {0,1} vs {2,3}
    const float rs_l = rs[row];
    const float e_l  = e[row];
    const float rc_l = rc[row];
    const int   type_c = (row < A_) ? atom_types[row] : -1;  // -1 never matches Z

    v8f acc = {0.0f, 0.0f, 0.0f, 0.0f, 0.0f, 0.0f, 0.0f, 0.0f};

#pragma unroll
    for (int c = 0; c < 16; ++c) {
        const int mA = c * 4 + hi * 2;       // this lane's two K (=m) values
        const float R0 = sR[w][mA];
        const float R1 = sR[w][mA + 1];
        const int   z0 = sZ[w][mA];
        const int   z1 = sZ[w][mA + 1];

        const float d0 = R0 - rs_l;
        const float d1 = R1 - rs_l;
        float a0 = __expf(-e_l * d0 * d0);
        float a1 = __expf(-e_l * d1 * d1);
        const float f0 = (R0 <= rc_l) ? 0.5f * (__cosf(PI_F * R0 / rc_l) + 1.0f) : 0.0f;
        const float f1 = (R1 <= rc_l) ? 0.5f * (__cosf(PI_F * R1 / rc_l) + 1.0f) : 0.0f;
        a0 *= f0;
        a1 *= f1;

        v2f av; av.x = a0; av.y = a1;
        v2f bv; bv.x = (z0 == type_c) ? 1.0f : 0.0f;
                bv.y = (z1 == type_c) ? 1.0f : 0.0f;

        // D = A(16x4, f32) x B(4x16, f32) + C   -> v_wmma_f32_16x16x4_f32
        acc = __builtin_amdgcn_wmma_f32_16x16x4_f32(
                  false, av, false, bv, (short)0, acc, false, false);
    }

    // ---- Phase 3: store. D layout: VGPR j = row (j + 8*hi), col = lane&15 ----
    if (row < A_) {
        const size_t bn = (size_t)b * N_ + n;
#pragma unroll
        for (int j = 0; j < 8; ++j) {
            const int l = j + 8 * hi;
            out[(size_t)(l * A_ + row) * (size_t)(B_ * N_) + bn] = acc[j];
        }
    }
}

extern "C" void kernel_launch(void* const* d_in, const int* in_sizes, int n_in,
                              void* d_out, int out_size, void* d_ws, size_t ws_size,
                              hipStream_t stream) {
    const float* X     = (const float*)d_in[0];
    const int*   Nbrs  = (const int*)d_in[1];
    const int*   NbrsZ = (const int*)d_in[2];
    const float* rc    = (const float*)d_in[3];
    const float* rs    = (const float*)d_in[4];
    const float* e     = (const float*)d_in[5];
    const int*   at    = (const int*)d_in[6];
    float*       out   = (float*)d_out;

    const int total_waves = B_ * N_;                    // 65536
    const int blocks = total_waves / WAVES_PER_BLOCK;   // 8192
    atomic_conv_wmma<<<blocks, WAVES_PER_BLOCK * 32, 0, stream>>>(
        X, Nbrs, NbrsZ, rc, rs, e, at, out);
}